// GAD_26302379721436
// MI455X (gfx1250) — compile-verified
//
#include <hip/hip_runtime.h>
#include <math.h>

// Problem constants (match reference)
#define Bb   128
#define Nn   512
#define Ff   64
#define Dd   64
#define Kk   20
#define BNt  (Bb*Nn)            // 65536 rows

typedef float v2f __attribute__((ext_vector_type(2)));
typedef float v8f __attribute__((ext_vector_type(8)));

// ---------------------------------------------------------------------------
// 1) embed row-normalize + embedding attention scores (per node, once)
//    e_norm[n] = embed[n] / (||embed[n]|| + 1e-12)
//    es_i[n]   = embed[n] . att_em_i ; es_j[n] = embed[n] . att_em_j
// ---------------------------------------------------------------------------
__global__ void k_embed_prep(const float* __restrict__ embed,
                             const float* __restrict__ att_em_i,
                             const float* __restrict__ att_em_j,
                             float* __restrict__ e_norm,
                             float* __restrict__ es_i,
                             float* __restrict__ es_j) {
  __shared__ float red[64];
  int n = blockIdx.x, d = threadIdx.x;
  float v = embed[n*Dd + d];

  red[d] = v*v; __syncthreads();
  for (int s = 32; s > 0; s >>= 1) { if (d < s) red[d] += red[d+s]; __syncthreads(); }
  float nrm = sqrtf(red[0]) + 1e-12f;
  __syncthreads();
  e_norm[n*Dd + d] = v / nrm;

  red[d] = v * att_em_i[d]; __syncthreads();
  for (int s = 32; s > 0; s >>= 1) { if (d < s) red[d] += red[d+s]; __syncthreads(); }
  if (d == 0) es_i[n] = red[0];
  __syncthreads();

  red[d] = v * att_em_j[d]; __syncthreads();
  for (int s = 32; s > 0; s >>= 1) { if (d < s) red[d] += red[d+s]; __syncthreads(); }
  if (d == 0) es_j[n] = red[0];
}

// ---------------------------------------------------------------------------
// 2) cosine-similarity row + iterative top-K argmax (tie-break: lower index,
//    matching jax.lax.top_k stability). One block per destination node.
// ---------------------------------------------------------------------------
__global__ void k_topk(const float* __restrict__ e_norm,
                       int* __restrict__ topk) {
  __shared__ float myrow[Dd];
  __shared__ float sim[Nn];
  __shared__ float rv[256];
  __shared__ int   ri[256];
  int n = blockIdx.x, t = threadIdx.x;
  if (t < Dd) myrow[t] = e_norm[n*Dd + t];
  __syncthreads();
  for (int m = t; m < Nn; m += 256) {
    const float* r = e_norm + m*Dd;
    float acc = 0.f;
    #pragma unroll
    for (int d = 0; d < Dd; d++) acc += myrow[d] * r[d];
    sim[m] = acc;
  }
  __syncthreads();
  for (int k = 0; k < Kk; k++) {
    float v0 = sim[t], v1 = sim[t + 256];
    float bv; int bi;
    if (v1 > v0) { bv = v1; bi = t + 256; } else { bv = v0; bi = t; }
    rv[t] = bv; ri[t] = bi; __syncthreads();
    for (int s = 128; s > 0; s >>= 1) {
      if (t < s) {
        float ov = rv[t+s]; int oi = ri[t+s];
        if (ov > rv[t] || (ov == rv[t] && oi < ri[t])) { rv[t] = ov; ri[t] = oi; }
      }
      __syncthreads();
    }
    if (t == 0) { topk[n*Kk + k] = ri[0]; sim[ri[0]] = -__builtin_inff(); }
    __syncthreads();
  }
}

// ---------------------------------------------------------------------------
// 3) g = x @ W_lin  via V_WMMA_F32_16X16X4_F32 (fp32-exact, wave32).
//    One wave owns a 16-row M-tile; loops 4 N-tiles; K-loop = 16 steps of k=4.
//    A 16x4 layout: lanes0-15 M=lane K={4s,4s+1}; lanes16-31 M=lane-16 K={4s+2,4s+3}
//    B 4x16 layout: vgpr j, lane L: K = 4s + j + 2*(L>=16), N = L%16
//    D 16x16:       vgpr r, lane L: M = r + 8*(L>=16),      N = L%16
//    W is held TRANSPOSED in LDS (Wt[col][k], row stride padded to 66 floats)
//    so each lane's B fragment {W[4s+2hi][col], W[4s+2hi+1][col]} is one
//    contiguous 8-byte ds_load_b64 with a bank-conflict-free 264B lane stride.
// ---------------------------------------------------------------------------
#define WT_STRIDE 66

__global__ void k_gemm_lin(const float* __restrict__ x,
                           const float* __restrict__ W,
                           float* __restrict__ g) {
  __shared__ float Wt[Dd * WT_STRIDE];
  int t = threadIdx.x;
  for (int i = t; i < Ff*Dd; i += 256) {
    int r = i >> 6, c = i & 63;          // W[r][c], row-major in memory
    Wt[c*WT_STRIDE + r] = W[i];          // transposed + padded in LDS
  }
  __syncthreads();

  int wave = t >> 5, lane = t & 31;
  int row0 = (blockIdx.x * 8 + wave) * 16;
  int mr = lane & 15, hi = lane >> 4;

  v2f a[16];
  const float* xr = x + (size_t)(row0 + mr) * Ff + 2*hi;
  #pragma unroll
  for (int s = 0; s < 16; s++) { a[s].x = xr[4*s]; a[s].y = xr[4*s + 1]; }

  #pragma unroll
  for (int nt = 0; nt < 4; nt++) {
    int col0 = nt * 16;
    const float* wcol = &Wt[(col0 + mr) * WT_STRIDE + 2*hi];
    v8f c = {0.f,0.f,0.f,0.f,0.f,0.f,0.f,0.f};
    #pragma unroll
    for (int s = 0; s < 16; s++) {
      v2f b = *(const v2f*)(wcol + 4*s);   // 8B-aligned single LDS load
      c = __builtin_amdgcn_wmma_f32_16x16x4_f32(false, a[s], false, b,
                                                (short)0, c, false, false);
    }
    float* gp = g + (size_t)(row0 + 8*hi) * Dd + col0 + mr;
    #pragma unroll
    for (int r = 0; r < 8; r++) gp[r*Dd] = c[r];
  }
}

// ---------------------------------------------------------------------------
// 4) per-node scores: s_i = g.att_i + es_i[n], s_j = g.att_j + es_j[n]
// ---------------------------------------------------------------------------
__global__ void k_scores(const float* __restrict__ g,
                         const float* __restrict__ att_i,
                         const float* __restrict__ att_j,
                         const float* __restrict__ es_i,
                         const float* __restrict__ es_j,
                         float* __restrict__ s_i,
                         float* __restrict__ s_j) {
  int bn = blockIdx.x * 256 + threadIdx.x;
  int n = bn & (Nn - 1);
  const float4* gr = (const float4*)(g + (size_t)bn * Dd);
  const float4* ai = (const float4*)att_i;
  const float4* aj = (const float4*)att_j;
  float si = 0.f, sj = 0.f;
  #pragma unroll
  for (int i = 0; i < 16; i++) {
    float4 gv = gr[i], a = ai[i], b = aj[i];
    si += gv.x*a.x + gv.y*a.y + gv.z*a.z + gv.w*a.w;
    sj += gv.x*b.x + gv.y*b.y + gv.z*b.z + gv.w*b.w;
  }
  s_i[bn] = si + es_i[n];
  s_j[bn] = sj + es_j[n];
}

// ---------------------------------------------------------------------------
// 5) edge attention softmax (21 edges: K top-k, self masked -inf, + self loop)
//    + weighted aggregation. One block per destination row, 64 threads = D.
// ---------------------------------------------------------------------------
__global__ void k_attn_agg(const float* __restrict__ g,
                           const int* __restrict__ topk,
                           const float* __restrict__ s_i,
                           const float* __restrict__ s_j,
                           const float* __restrict__ bias,
                           float* __restrict__ agg) {
  __shared__ float w[Kk + 1];
  __shared__ int   srcs[Kk + 1];
  int bn = blockIdx.x;
  int b  = bn >> 9;        // / Nn
  int n  = bn & (Nn - 1);
  int t  = threadIdx.x;

  if (t < Kk + 1) {
    int src; float l;
    if (t < Kk) {
      src = topk[n*Kk + t];
      if (src == n) {
        l = -__builtin_inff();             // remove_self_loops mask
      } else {
        float z = s_i[bn] + s_j[b*Nn + src];
        l = z > 0.f ? z : 0.2f * z;        // leaky_relu(0.2)
      }
    } else {
      src = n;                             // added self loop (never masked)
      float z = s_i[bn] + s_j[bn];
      l = z > 0.f ? z : 0.2f * z;
    }
    srcs[t] = src; w[t] = l;
  }
  __syncthreads();
  if (t == 0) {
    float m = -__builtin_inff();
    for (int e = 0; e <= Kk; e++) m = fmaxf(m, w[e]);
    float ex[Kk + 1];
    float den = 0.f;
    for (int e = 0; e <= Kk; e++) { ex[e] = expf(w[e] - m); den += ex[e]; }
    float invd = 1.f / den;
    for (int e = 0; e <= Kk; e++) w[e] = ex[e] * invd;
  }
  __syncthreads();

  float acc = bias[t];
  size_t rowb = (size_t)b * Nn;
  for (int e = 0; e <= Kk; e++) {
    float we = w[e];
    if (we != 0.f) acc += we * g[(rowb + srcs[e]) * Dd + t];
  }
  agg[(size_t)bn * Dd + t] = acc;
}

// ---------------------------------------------------------------------------
// 6) BN1 channel stats over 65536 rows: deterministic two-stage reduction.
//    Thread channel = tid & 63 (grid stride is a multiple of 64).
// ---------------------------------------------------------------------------
__global__ void k_bn1_part(const float* __restrict__ agg,
                           float* __restrict__ psum, float* __restrict__ psq) {
  __shared__ float ss[256], sq[256];
  int t = threadIdx.x;
  size_t stride = (size_t)gridDim.x * 256;
  float a = 0.f, q = 0.f;
  for (size_t idx = (size_t)blockIdx.x * 256 + t; idx < (size_t)BNt * Dd; idx += stride) {
    float v = agg[idx]; a += v; q += v*v;
  }
  ss[t] = a; sq[t] = q; __syncthreads();
  if (t < 64) {
    a = ss[t] + ss[t+64] + ss[t+128] + ss[t+192];
    q = sq[t] + sq[t+64] + sq[t+128] + sq[t+192];
    psum[blockIdx.x*64 + t] = a; psq[blockIdx.x*64 + t] = q;
  }
}

// BN2 stats over t = relu(bn1(agg)) * embed[n], computed on the fly
__global__ void k_bn2_part(const float* __restrict__ agg,
                           const float* __restrict__ embed,
                           const float* __restrict__ sum1, const float* __restrict__ sq1,
                           const float* __restrict__ g1, const float* __restrict__ b1,
                           float* __restrict__ psum, float* __restrict__ psq) {
  __shared__ float ss[256], ssq[256];
  int t = threadIdx.x;
  int c = t & 63;
  const float inv = 1.f / (float)BNt;
  float mu  = sum1[c] * inv;
  float var = sq1[c] * inv - mu*mu;
  float iv  = rsqrtf(var + 1e-5f);
  float ga = g1[c], be = b1[c];
  size_t stride = (size_t)gridDim.x * 256;
  float a = 0.f, q = 0.f;
  for (size_t idx = (size_t)blockIdx.x * 256 + t; idx < (size_t)BNt * Dd; idx += stride) {
    float v = agg[idx];
    float h = (v - mu) * iv * ga + be;
    h = h > 0.f ? h : 0.f;
    int n = (int)((idx >> 6) & (Nn - 1));
    h *= embed[n*Dd + c];
    a += h; q += h*h;
  }
  ss[t] = a; ssq[t] = q; __syncthreads();
  if (t < 64) {
    a = ss[t] + ss[t+64] + ss[t+128] + ss[t+192];
    q = ssq[t] + ssq[t+64] + ssq[t+128] + ssq[t+192];
    psum[blockIdx.x*64 + t] = a; psq[blockIdx.x*64 + t] = q;
  }
}

__global__ void k_reduce_stats(const float* __restrict__ psum,
                               const float* __restrict__ psq,
                               int nblk,
                               float* __restrict__ osum, float* __restrict__ osq) {
  int c = threadIdx.x;
  float a = 0.f, q = 0.f;
  for (int i = 0; i < nblk; i++) { a += psum[i*64 + c]; q += psq[i*64 + c]; }
  osum[c] = a; osq[c] = q;
}

// ---------------------------------------------------------------------------
// 7) final: recompute t = relu(bn1)*embed, apply bn2 + relu, project to 1
// ---------------------------------------------------------------------------
__global__ void k_final(const float* __restrict__ agg,
                        const float* __restrict__ embed,
                        const float* __restrict__ sum1, const float* __restrict__ sq1,
                        const float* __restrict__ sum2, const float* __restrict__ sq2,
                        const float* __restrict__ g1, const float* __restrict__ b1,
                        const float* __restrict__ g2, const float* __restrict__ b2,
                        const float* __restrict__ outW, const float* __restrict__ outB,
                        float* __restrict__ out) {
  __shared__ float red[64];
  int bn = blockIdx.x, c = threadIdx.x, n = bn & (Nn - 1);
  const float inv = 1.f / (float)BNt;
  float mu1 = sum1[c]*inv, v1 = sq1[c]*inv - mu1*mu1, iv1 = rsqrtf(v1 + 1e-5f);
  float mu2 = sum2[c]*inv, v2 = sq2[c]*inv - mu2*mu2, iv2 = rsqrtf(v2 + 1e-5f);
  float h = (agg[(size_t)bn*Dd + c] - mu1) * iv1 * g1[c] + b1[c];
  h = h > 0.f ? h : 0.f;
  h *= embed[n*Dd + c];
  float y = (h - mu2) * iv2 * g2[c] + b2[c];
  y = y > 0.f ? y : 0.f;
  red[c] = y * outW[c]; __syncthreads();
  for (int s = 32; s > 0; s >>= 1) { if (c < s) red[c] += red[c+s]; __syncthreads(); }
  if (c == 0) out[bn] = red[0] + outB[0];
}

// ---------------------------------------------------------------------------
extern "C" void kernel_launch(void* const* d_in, const int* in_sizes, int n_in,
                              void* d_out, int out_size, void* d_ws, size_t ws_size,
                              hipStream_t stream) {
  const float* data     = (const float*)d_in[0];
  /* d_in[1] org_edge_index: unused by the reference computation */
  const float* embed    = (const float*)d_in[2];
  const float* W_lin    = (const float*)d_in[3];
  const float* att_i    = (const float*)d_in[4];
  const float* att_j    = (const float*)d_in[5];
  const float* att_em_i = (const float*)d_in[6];
  const float* att_em_j = (const float*)d_in[7];
  const float* gnn_bias = (const float*)d_in[8];
  const float* bn1_g    = (const float*)d_in[9];
  const float* bn1_b    = (const float*)d_in[10];
  const float* bn2_g    = (const float*)d_in[11];
  const float* bn2_b    = (const float*)d_in[12];
  const float* out_W    = (const float*)d_in[13];
  const float* out_b    = (const float*)d_in[14];
  float* out = (float*)d_out;
  (void)in_sizes; (void)n_in; (void)out_size; (void)ws_size;

  char* base = (char*)d_ws;
  size_t off = 0;
  auto take = [&](size_t bytes) -> char* {
    char* p = base + off;
    off += (bytes + 255) & ~(size_t)255;
    return p;
  };
  float* e_norm = (float*)take((size_t)Nn*Dd*sizeof(float));
  int*   topk   = (int*)  take((size_t)Nn*Kk*sizeof(int));
  float* es_i   = (float*)take(Nn*sizeof(float));
  float* es_j   = (float*)take(Nn*sizeof(float));
  float* g      = (float*)take((size_t)BNt*Dd*sizeof(float));   // 16 MB
  float* s_i    = (float*)take((size_t)BNt*sizeof(float));
  float* s_j    = (float*)take((size_t)BNt*sizeof(float));
  float* agg    = (float*)take((size_t)BNt*Dd*sizeof(float));   // 16 MB
  float* p1s    = (float*)take(512*64*sizeof(float));
  float* p1q    = (float*)take(512*64*sizeof(float));
  float* p2s    = (float*)take(512*64*sizeof(float));
  float* p2q    = (float*)take(512*64*sizeof(float));
  float* sum1   = (float*)take(64*sizeof(float));
  float* sq1    = (float*)take(64*sizeof(float));
  float* sum2   = (float*)take(64*sizeof(float));
  float* sq2    = (float*)take(64*sizeof(float));

  k_embed_prep<<<Nn, 64, 0, stream>>>(embed, att_em_i, att_em_j, e_norm, es_i, es_j);
  k_topk      <<<Nn, 256, 0, stream>>>(e_norm, topk);
  k_gemm_lin  <<<BNt/128, 256, 0, stream>>>(data, W_lin, g);      // 8 waves/block, 16-row tile/wave
  k_scores    <<<BNt/256, 256, 0, stream>>>(g, att_i, att_j, es_i, es_j, s_i, s_j);
  k_attn_agg  <<<BNt, 64, 0, stream>>>(g, topk, s_i, s_j, gnn_bias, agg);
  k_bn1_part  <<<512, 256, 0, stream>>>(agg, p1s, p1q);
  k_reduce_stats<<<1, 64, 0, stream>>>(p1s, p1q, 512, sum1, sq1);
  k_bn2_part  <<<512, 256, 0, stream>>>(agg, embed, sum1, sq1, bn1_g, bn1_b, p2s, p2q);
  k_reduce_stats<<<1, 64, 0, stream>>>(p2s, p2q, 512, sum2, sq2);
  k_final     <<<BNt, 64, 0, stream>>>(agg, embed, sum1, sq1, sum2, sq2,
                                       bn1_g, bn1_b, bn2_g, bn2_b, out_W, out_b, out);
}